// PointNetSetAbstraction_69080253989614
// MI455X (gfx1250) — compile-verified
//
#include <hip/hip_runtime.h>

// ---------------------------------------------------------------------------
// MI455X (gfx1250) fused DGCNN set-abstraction.
// B=4, N=4096, K=16, MLP=[64,64,128].  All fp32; matrix work on
// V_WMMA_F32_16X16X4_F32 (M=16 rows == the 16 kNN neighbors per point).
// ---------------------------------------------------------------------------

typedef float v2f __attribute__((ext_vector_type(2)));
typedef float v8f __attribute__((ext_vector_type(8)));

#define BB   4
#define NN   4096
#define KNB  16
#define LRELU 0.2f
// 1/sqrt(1+1e-5)  (BatchNorm eval with mean=0, var=1)
#define BN_INV 0.9999950000374996f

__device__ __forceinline__ v8f wmma4(v2f a, v2f b, v8f c) {
  // D = A(16x4,f32) * B(4x16,f32) + C(16x16,f32)
  return __builtin_amdgcn_wmma_f32_16x16x4_f32(
      /*neg_a=*/false, a, /*neg_b=*/false, b,
      /*c_mod=*/(short)0, c, /*reuse_a=*/false, /*reuse_b=*/false);
}

// ---------------------------------------------------------------------------
// kNN (top-16 smallest squared distance, self included, ties -> lower index).
// One thread per query; query row in registers; candidate tiles in LDS.
// ---------------------------------------------------------------------------
template <int C>
__global__ __launch_bounds__(64) void knn_kernel(
    const float* __restrict__ X, int ldX, int* __restrict__ idxOut) {
  constexpr int TJ = 64;
  __shared__ float tile[TJ][C];

  const int blocksPerB = NN / TJ;
  const int b    = blockIdx.x / blocksPerB;
  const int iblk = blockIdx.x % blocksPerB;
  const int i    = iblk * TJ + threadIdx.x;
  const float* Xb = X + (size_t)b * NN * ldX;

  float q[C];
#pragma unroll
  for (int c = 0; c < C; ++c) q[c] = Xb[(size_t)i * ldX + c];

  float bd[KNB];
  int   bi[KNB];
#pragma unroll
  for (int t = 0; t < KNB; ++t) { bd[t] = 3.4e38f; bi[t] = 0; }

  for (int j0 = 0; j0 < NN; j0 += TJ) {
    __syncthreads();
    for (int t = threadIdx.x; t < TJ * C; t += 64) {
      int jj = t / C, c = t % C;
      tile[jj][c] = Xb[(size_t)(j0 + jj) * ldX + c];
    }
    __syncthreads();
    for (int jj = 0; jj < TJ; ++jj) {
      float d = 0.f;
#pragma unroll
      for (int c = 0; c < C; ++c) {
        float df = q[c] - tile[jj][c];
        d = fmaf(df, df, d);
      }
      if (d < bd[KNB - 1]) {
        float cd = d;
        int   ci = j0 + jj;
#pragma unroll
        for (int t = 0; t < KNB; ++t) {   // sorted insert via swap chain
          bool  sw = cd < bd[t];
          float td = bd[t];
          int   ti = bi[t];
          bd[t] = sw ? cd : td;
          bi[t] = sw ? ci : ti;
          cd = sw ? td : cd;
          ci = sw ? ti : ci;
        }
      }
    }
  }
#pragma unroll
  for (int t = 0; t < KNB; ++t)
    idxOut[((size_t)(b * NN + i)) * KNB + t] = bi[t];
}

// ---------------------------------------------------------------------------
// Position encoding: xyz1[b,n,k,10] = [dist, rel(3), tile(3), neigh(3)]
// ---------------------------------------------------------------------------
__global__ void posenc_kernel(const float* __restrict__ xyz,
                              const int* __restrict__ idx,
                              float* __restrict__ xyz1) {
  int t = blockIdx.x * blockDim.x + threadIdx.x;   // over B*N*K
  if (t >= BB * NN * KNB) return;
  int pn = t >> 4;           // b*N + n
  int b  = pn / NN;
  int j  = idx[t];
  const float* pc = xyz + (size_t)pn * 3;
  const float* pj = xyz + ((size_t)(b * NN + j)) * 3;
  float tx = pc[0], ty = pc[1], tz = pc[2];
  float nx = pj[0], ny = pj[1], nz = pj[2];
  float rx = tx - nx, ry = ty - ny, rz = tz - nz;
  float dist = sqrtf(rx * rx + ry * ry + rz * rz);
  float* o = xyz1 + (size_t)t * 10;
  o[0] = dist;
  o[1] = rx; o[2] = ry; o[3] = rz;
  o[4] = tx; o[5] = ty; o[6] = tz;
  o[7] = nx; o[8] = ny; o[9] = nz;
}

// ---------------------------------------------------------------------------
// Fused layer: edge_conv (WMMA) -> concat xyz1 -> l1 logits (WMMA) ->
// softmax over k -> weighted agg -> output 1x1 conv + BN + lrelu (+resid).
// One block (4 wave32) per point; the 16 neighbors are the WMMA M dimension.
// ---------------------------------------------------------------------------
template <int CIN_, int COUT, int DD>
__global__ __launch_bounds__(128) void fused_layer_kernel(
    const float* __restrict__ Fprev, int ldF,
    const int* __restrict__ idx,
    const float* __restrict__ xyz1,
    const float* __restrict__ ecW, const float* __restrict__ ecb,
    const float* __restrict__ ecg, const float* __restrict__ ecbt,
    const float* __restrict__ l1W, const float* __restrict__ l1b,
    const float* __restrict__ cW,  const float* __restrict__ cb,
    const float* __restrict__ cg,  const float* __restrict__ cbt,
    float* __restrict__ Out, int ldOut, int colOff,
    const float* __restrict__ resid) {
  constexpr int KEC = 2 * CIN_;            // edge-conv input channels
  constexpr int NT1 = COUT / 16;           // edge-conv col tiles
  constexpr int NT2 = (DD + 15) / 16;      // logits col tiles (padded)

  __shared__ float cen[CIN_];
  __shared__ float nb[KNB][CIN_];
  __shared__ float x1[KNB][10];
  __shared__ float npL[KNB][COUT];
  __shared__ float lgL[KNB][NT2 * 16];
  __shared__ float aggL[DD];

  const int pt  = blockIdx.x;              // b*N + n
  const int tid = threadIdx.x;
  const int b   = pt / NN;

  // ---- stage central row, 16 neighbor rows, and xyz1 rows into LDS ----
  for (int t = tid; t < CIN_; t += 128) cen[t] = Fprev[(size_t)pt * ldF + t];
  for (int t = tid; t < KNB * CIN_; t += 128) {
    int k = t / CIN_, c = t % CIN_;
    int j = idx[pt * KNB + k];
    nb[k][c] = Fprev[((size_t)(b * NN + j)) * ldF + c];
  }
  for (int t = tid; t < KNB * 10; t += 128)
    x1[t / 10][t % 10] = xyz1[(size_t)pt * (KNB * 10) + t];
  __syncthreads();

  const int lane = tid & 31;
  const int wv   = tid >> 5;        // wave 0..3
  const int half = lane >> 4;       // lane-half selects K pair / M offset
  const int l16  = lane & 15;

  // ---- GEMM1: np[16][COUT] = lrelu(bn(ef * ecW + ecb)) ----
  for (int tile = wv; tile < NT1; tile += 4) {
    v8f acc = {0.f, 0.f, 0.f, 0.f, 0.f, 0.f, 0.f, 0.f};
    const int colB = tile * 16 + l16;
    for (int k0 = 0; k0 < KEC; k0 += 4) {
      const int ka = k0 + 2 * half;
      const int k1 = ka + 1;
      int kaC = ka < KEC ? ka : KEC - 1;
      int k1C = k1 < KEC ? k1 : KEC - 1;
      float a0 = (kaC < CIN_) ? cen[kaC] : (nb[l16][kaC - CIN_] - cen[kaC - CIN_]);
      float a1 = (k1C < CIN_) ? cen[k1C] : (nb[l16][k1C - CIN_] - cen[k1C - CIN_]);
      a0 = (ka < KEC) ? a0 : 0.f;
      a1 = (k1 < KEC) ? a1 : 0.f;
      float b0 = (ka < KEC) ? ecW[kaC * COUT + colB] : 0.f;
      float b1 = (k1 < KEC) ? ecW[k1C * COUT + colB] : 0.f;
      v2f av; av[0] = a0; av[1] = a1;
      v2f bv; bv[0] = b0; bv[1] = b1;
      acc = wmma4(av, bv, acc);
    }
    const float g  = ecg[colB];
    const float bt = ecbt[colB];
    const float bs = ecb[colB];
#pragma unroll
    for (int r = 0; r < 8; ++r) {
      int   m = r + 8 * half;
      float v = acc[r] + bs;
      v = g * (v * BN_INV) + bt;
      v = v > 0.f ? v : LRELU * v;
      npL[m][tile * 16 + l16] = v;
    }
  }
  __syncthreads();

  // ---- GEMM2: logits[16][DD] = [np | xyz1] * l1W + l1b ----
  for (int tile = wv; tile < NT2; tile += 4) {
    v8f acc = {0.f, 0.f, 0.f, 0.f, 0.f, 0.f, 0.f, 0.f};
    const int colB = tile * 16 + l16;
    const int colC = colB < DD ? colB : DD - 1;
    for (int k0 = 0; k0 < DD; k0 += 4) {
      const int ka = k0 + 2 * half;
      const int k1 = ka + 1;
      int kaC = ka < DD ? ka : DD - 1;
      int k1C = k1 < DD ? k1 : DD - 1;
      float a0 = (kaC < COUT) ? npL[l16][kaC] : x1[l16][kaC - COUT];
      float a1 = (k1C < COUT) ? npL[l16][k1C] : x1[l16][k1C - COUT];
      a0 = (ka < DD) ? a0 : 0.f;
      a1 = (k1 < DD) ? a1 : 0.f;
      float b0 = (ka < DD && colB < DD) ? l1W[kaC * DD + colC] : 0.f;
      float b1 = (k1 < DD && colB < DD) ? l1W[k1C * DD + colC] : 0.f;
      v2f av; av[0] = a0; av[1] = a1;
      v2f bv; bv[0] = b0; bv[1] = b1;
      acc = wmma4(av, bv, acc);
    }
    const float bs = (colB < DD) ? l1b[colB] : 0.f;
#pragma unroll
    for (int r = 0; r < 8; ++r) {
      int m = r + 8 * half;
      lgL[m][tile * 16 + l16] = acc[r] + bs;
    }
  }
  __syncthreads();

  // ---- softmax over k (axis=2) per channel + weighted aggregation ----
  for (int d = tid; d < DD; d += 128) {
    float mx = -3.4e38f;
#pragma unroll
    for (int k = 0; k < KNB; ++k) mx = fmaxf(mx, lgL[k][d]);
    float s = 0.f, wsum = 0.f;
#pragma unroll
    for (int k = 0; k < KNB; ++k) {
      float e = __expf(lgL[k][d] - mx);
      float f = (d < COUT) ? npL[k][d] : x1[k][d - COUT];
      s += e;
      wsum = fmaf(f, e, wsum);
    }
    aggL[d] = wsum / s;
  }
  __syncthreads();

  // ---- output GEMV: out[c] = lrelu(bn(agg . cW[:,c] + cb)) (+resid) ----
  for (int c = tid; c < COUT; c += 128) {
    float v = 0.f;
    for (int d = 0; d < DD; ++d) v = fmaf(aggL[d], cW[d * COUT + c], v);
    v += cb[c];
    v = cg[c] * (v * BN_INV) + cbt[c];
    v = v > 0.f ? v : LRELU * v;
    if (resid) v += resid[(size_t)pt * ldOut + c];
    Out[(size_t)pt * ldOut + colOff + c] = v;
  }
}

// ---------------------------------------------------------------------------
// Host-side orchestration (graph-capture safe: only kernel launches).
// ---------------------------------------------------------------------------
extern "C" void kernel_launch(void* const* d_in, const int* in_sizes, int n_in,
                              void* d_out, int out_size, void* d_ws,
                              size_t ws_size, hipStream_t stream) {
  (void)in_sizes; (void)n_in; (void)out_size; (void)ws_size;

  const float* xyz = (const float*)d_in[0];   // new_xyz  [B,N,3]
  const float* pts = (const float*)d_in[1];   // new_points [B,N,3]
  auto P = [&](int layer, int p) -> const float* {
    return (const float*)d_in[2 + layer * 10 + p];
  };
  // param order per layer: 0 ecW 1 ecb 2 ecg 3 ecbt 4 l1W 5 l1b 6 cW 7 cb 8 g 9 bt

  // workspace layout
  char* w = (char*)d_ws;
  int* idx0 = (int*)w;  w += (size_t)BB * NN * KNB * sizeof(int);
  int* idx1 = (int*)w;  w += (size_t)BB * NN * KNB * sizeof(int);
  int* idx2 = (int*)w;  w += (size_t)BB * NN * KNB * sizeof(int);
  float* xyz1  = (float*)w; w += (size_t)BB * NN * KNB * 10 * sizeof(float);
  float* f2buf = (float*)w; w += (size_t)BB * NN * 128 * sizeof(float);

  const dim3 knnGrid(BB * (NN / 64)), knnBlk(64);
  const dim3 ptGrid(BB * NN), ptBlk(128);

  // layer 0: kNN on raw xyz (C=3), shared with position encoding
  knn_kernel<3><<<knnGrid, knnBlk, 0, stream>>>(pts, 3, idx0);
  {
    int tot = BB * NN * KNB;
    posenc_kernel<<<(tot + 255) / 256, 256, 0, stream>>>(xyz, idx0, xyz1);
  }
  // f1 -> columns [0,64) of f2buf (ld=128) so f2 = [f1|f2a] needs no copy
  fused_layer_kernel<3, 64, 74><<<ptGrid, ptBlk, 0, stream>>>(
      pts, 3, idx0, xyz1,
      P(0,0), P(0,1), P(0,2), P(0,3), P(0,4), P(0,5), P(0,6), P(0,7), P(0,8), P(0,9),
      f2buf, 128, 0, nullptr);

  // layer 1: kNN on f1 (C=64, ld=128), write f2a -> columns [64,128)
  knn_kernel<64><<<knnGrid, knnBlk, 0, stream>>>(f2buf, 128, idx1);
  fused_layer_kernel<64, 64, 74><<<ptGrid, ptBlk, 0, stream>>>(
      f2buf, 128, idx1, xyz1,
      P(1,0), P(1,1), P(1,2), P(1,3), P(1,4), P(1,5), P(1,6), P(1,7), P(1,8), P(1,9),
      f2buf, 128, 64, nullptr);

  // layer 2: kNN on f2 (C=128); output = f3 + f2 (residual) straight to d_out
  knn_kernel<128><<<knnGrid, knnBlk, 0, stream>>>(f2buf, 128, idx2);
  fused_layer_kernel<128, 128, 138><<<ptGrid, ptBlk, 0, stream>>>(
      f2buf, 128, idx2, xyz1,
      P(2,0), P(2,1), P(2,2), P(2,3), P(2,4), P(2,5), P(2,6), P(2,7), P(2,8), P(2,9),
      (float*)d_out, 128, 0, f2buf);
}